// VectorQuantizer_515396076132
// MI455X (gfx1250) — compile-verified
//
#include <hip/hip_runtime.h>
#include <hip/hip_bf16.h>
#include <math.h>

typedef __attribute__((ext_vector_type(2))) float v2f;
typedef __attribute__((ext_vector_type(8))) float v8f;

#define NE      1024
#define EDIM    64
#define NB      32
#define HW      4096      // 64*64 spatial per batch
#define NROWS   131072    // NB*HW
#define APAD    68        // padded LDS row stride (conflict-free for the frag pattern)

// d_out layout (flat f32, tuple order): loss, z_q, perplexity, min_encodings, min_encoding_indices
#define ZQ_OFF   ((size_t)1)
#define PPL_OFF  ((size_t)8388609)
#define ENC_OFF  ((size_t)8388610)
#define IDX_OFF  ((size_t)142606338)
#define ENC_N    ((size_t)134217728)

#define WMMA_F32(a, b, c) \
    __builtin_amdgcn_wmma_f32_16x16x4_f32(false, (a), false, (b), (short)0, (c), false, false)

// ---------------- prep: embedding norms + zero scratch ----------------
__global__ void vq_prep(const float* __restrict__ emb, float* __restrict__ enorm,
                        unsigned int* __restrict__ hist, float* __restrict__ lacc) {
    int t = blockIdx.x * blockDim.x + threadIdx.x;   // 1024 threads
    if (t < NE) {
        const float* e = emb + (size_t)t * EDIM;
        float s = 0.f;
        #pragma unroll 8
        for (int c = 0; c < EDIM; ++c) s += e[c] * e[c];
        enorm[t] = s;
        hist[t] = 0u;
    }
    if (t == 0) *lacc = 0.f;
}

// ------- zero-fill one-hot region: b128 body + scalar head/tail (start is 8B-aligned) -------
__global__ void vq_fill(float* __restrict__ enc) {
    // body: float4s over enc[2 .. 2 + 4*n4)
    const size_t n4 = (ENC_N - 2) / 4;               // 33554431
    float4* p4 = (float4*)(enc + 2);
    size_t i = (size_t)blockIdx.x * blockDim.x + threadIdx.x;
    size_t stride = (size_t)gridDim.x * blockDim.x;
    float4 z4; z4.x = 0.f; z4.y = 0.f; z4.z = 0.f; z4.w = 0.f;
    for (; i < n4; i += stride) p4[i] = z4;
    if (blockIdx.x == 0 && threadIdx.x == 0) {
        enc[0] = 0.f; enc[1] = 0.f;
        enc[ENC_N - 2] = 0.f; enc[ENC_N - 1] = 0.f;
    }
}

// ---------------- main: WMMA f32 GEMM + argmin + outputs ----------------
// Block = 256 threads (8 waves). Each wave: 2 strips of 16 columns -> 256 cols/block.
// Embedding tiles (16x64) staged in double-buffered LDS, shared by all 8 waves.
__global__ __launch_bounds__(256) void vq_main(const float* __restrict__ z,
                                               const float* __restrict__ emb,
                                               const float* __restrict__ enorm,
                                               unsigned int* __restrict__ hist,
                                               float* __restrict__ lacc,
                                               float* __restrict__ out) {
    __shared__ float atile[2][16 * APAD];

    const int tid  = threadIdx.x;
    const int lane = tid & 31;
    const int wave = tid >> 5;
    const int b    = blockIdx.x >> 4;            // 16 blocks per batch
    const int cg   = blockIdx.x & 15;            // 256 cols per block
    const int n    = lane & 15;
    const int hi   = lane >> 4;                  // K/M half select
    const int koff = hi * 2;
    const int col0 = cg * 256 + wave * 32 + n;
    const int col1 = col0 + 16;

    const float* zb = z + (size_t)b * EDIM * HW;

    // B fragments for both strips: lane holds channels {4j+koff, 4j+koff+1}
    float bf0[32], bf1[32];
    #pragma unroll
    for (int j = 0; j < 16; ++j) {
        int c0 = 4 * j + koff;
        bf0[2*j]     = zb[(size_t)c0 * HW + col0];
        bf0[2*j + 1] = zb[(size_t)(c0 + 1) * HW + col0];
        bf1[2*j]     = zb[(size_t)c0 * HW + col1];
        bf1[2*j + 1] = zb[(size_t)(c0 + 1) * HW + col1];
    }

    // cooperative staging: thread handles 4 consecutive floats of the 1024-float tile
    const int sm = tid >> 4;                     // tile row 0..15
    const int sc = (tid & 15) * 4;               // tile col (16B aligned)
    *(float4*)&atile[0][sm * APAD + sc] = *(const float4*)(emb + (size_t)sm * EDIM + sc);
    __syncthreads();

    float minv0 = 3.4e38f, minv1 = 3.4e38f;
    int   mini0 = 0, mini1 = 0;

    for (int t = 0; t < 64; ++t) {
        const int p = t & 1;
        float4 nxt;
        if (t + 1 < 64)   // issue next tile's global load early; used after compute
            nxt = *(const float4*)(emb + (size_t)((t + 1) * 16 + sm) * EDIM + sc);

        const float* ap = &atile[p][n * APAD + koff];
        v8f acc00 = {0.f,0.f,0.f,0.f,0.f,0.f,0.f,0.f};
        v8f acc01 = {0.f,0.f,0.f,0.f,0.f,0.f,0.f,0.f};
        v8f acc10 = {0.f,0.f,0.f,0.f,0.f,0.f,0.f,0.f};
        v8f acc11 = {0.f,0.f,0.f,0.f,0.f,0.f,0.f,0.f};
        #pragma unroll
        for (int j = 0; j < 16; ++j) {
            v2f a;  a.x  = ap[4*j];      a.y  = ap[4*j + 1];      // ds_load_b64
            v2f b0; b0.x = bf0[2*j];     b0.y = bf0[2*j + 1];
            v2f b1; b1.x = bf1[2*j];     b1.y = bf1[2*j + 1];
            if (j & 1) { acc01 = WMMA_F32(a, b0, acc01); acc11 = WMMA_F32(a, b1, acc11); }
            else       { acc00 = WMMA_F32(a, b0, acc00); acc10 = WMMA_F32(a, b1, acc10); }
        }

        const float4 en0 = *(const float4*)(enorm + t * 16 + hi * 8);
        const float4 en1 = *(const float4*)(enorm + t * 16 + hi * 8 + 4);
        const float en[8] = {en0.x, en0.y, en0.z, en0.w, en1.x, en1.y, en1.z, en1.w};
        #pragma unroll
        for (int v = 0; v < 8; ++v) {
            const int mg = t * 16 + hi * 8 + v;
            float d0 = en[v] - 2.0f * (acc00[v] + acc01[v]);
            float d1 = en[v] - 2.0f * (acc10[v] + acc11[v]);
            if (d0 < minv0) { minv0 = d0; mini0 = mg; }
            if (d1 < minv1) { minv1 = d1; mini1 = mg; }
        }

        if (t + 1 < 64)
            *(float4*)&atile[1 - p][sm * APAD + sc] = nxt;
        __syncthreads();
    }

    // combine the lane pair holding the same column (M rows 0-7 / 8-15)
    float lsum = 0.f;
    #pragma unroll
    for (int s = 0; s < 2; ++s) {
        float mv = s ? minv1 : minv0;
        int   mi = s ? mini1 : mini0;
        const int col = s ? col1 : col0;
        float omv = __shfl_xor(mv, 16, 32);
        int   omi = __shfl_xor(mi, 16, 32);
        if (omv < mv || (omv == mv && omi < mi)) { mv = omv; mi = omi; }

        const int r = b * HW + col;
        if (hi == 0) {
            out[IDX_OFF + (size_t)r] = (float)mi;
            out[ENC_OFF + (size_t)r * NE + (size_t)mi] = 1.0f;
            atomicAdd(&hist[mi], 1u);
        }
        // z_q (NCHW) + loss; lane pair splits channels, bf* holds this lane's z values
        const float* eq = emb + (size_t)mi * EDIM;
        float* zq = out + ZQ_OFF + (size_t)b * EDIM * HW + col;
        const float* bf = s ? bf1 : bf0;
        #pragma unroll
        for (int j = 0; j < 16; ++j) {
            int c0 = 4 * j + koff;
            float e0 = eq[c0], e1 = eq[c0 + 1];
            zq[(size_t)c0 * HW]       = e0;
            zq[(size_t)(c0 + 1) * HW] = e1;
            float d0 = e0 - bf[2*j];
            float d1 = e1 - bf[2*j + 1];
            lsum += d0 * d0 + d1 * d1;
        }
    }
    #pragma unroll
    for (int s2 = 16; s2 > 0; s2 >>= 1) lsum += __shfl_xor(lsum, s2, 32);
    if (lane == 0) atomicAdd(lacc, lsum);
}

// ---------------- final: perplexity + loss scalars ----------------
__global__ void vq_final(const unsigned int* __restrict__ hist,
                         const float* __restrict__ lacc,
                         float* __restrict__ out) {
    __shared__ float sdata[1024];
    int t = threadIdx.x;
    float p = (float)hist[t] * (1.0f / (float)NROWS);
    sdata[t] = p * logf(p + 1e-10f);
    __syncthreads();
    for (int s = 512; s > 0; s >>= 1) {
        if (t < s) sdata[t] += sdata[t + s];
        __syncthreads();
    }
    if (t == 0) {
        out[PPL_OFF] = expf(-sdata[0]);
        out[0] = 1.25f * (*lacc) / 8388608.0f;   // (1 + BETA) * mean((z_q - z)^2)
    }
}

extern "C" void kernel_launch(void* const* d_in, const int* in_sizes, int n_in,
                              void* d_out, int out_size, void* d_ws, size_t ws_size,
                              hipStream_t stream) {
    const float* z   = (const float*)d_in[0];
    const float* emb = (const float*)d_in[1];
    float* out = (float*)d_out;

    float*        enorm = (float*)d_ws;                              // 4 KB
    unsigned int* hist  = (unsigned int*)((char*)d_ws + 4096);       // 4 KB
    float*        lacc  = (float*)((char*)d_ws + 8192);              // 4 B

    vq_prep<<<4, 256, 0, stream>>>(emb, enorm, hist, lacc);
    vq_fill<<<8192, 256, 0, stream>>>(out + ENC_OFF);
    vq_main<<<512, 256, 0, stream>>>(z, emb, enorm, hist, lacc, out);
    vq_final<<<1, 1024, 0, stream>>>(hist, lacc, out);
}